// TemporalGNN_76424648065502
// MI455X (gfx1250) — compile-verified
//
#include <hip/hip_runtime.h>
#include <hip/hip_bf16.h>
#include <math.h>
#include <stdint.h>

typedef float v2f __attribute__((ext_vector_type(2)));
typedef float v8f __attribute__((ext_vector_type(8)));

#define NN    83
#define NPAD  96
#define CC    128
#define FIN   32
#define TT    256
#define BB    32
#define HSZ   (BB * NPAD * CC)   // elements per [B,NPAD,C] tensor

// ---------------------------------------------------------------------------
// Async global->LDS copy of one 16x128 f32 tile (8 KB) using CDNA5
// GLOBAL_LOAD_ASYNC_TO_LDS_B128 (ASYNCcnt-tracked, bypasses VGPRs).
// 256 threads x 16B x 2 chunks. Generic LDS pointer truncates to LDS offset.
// ---------------------------------------------------------------------------
__device__ inline void async_tile_load(const float* __restrict__ gsrc,
                                       void* lds_dst, int tid) {
    uint32_t l = (uint32_t)(uintptr_t)lds_dst + (uint32_t)tid * 16u;
    int      voff = tid * 16;
    asm volatile("global_load_async_to_lds_b128 %0, %1, %2 offset:0"
                 :: "v"(l), "v"(voff), "s"(gsrc) : "memory");
    asm volatile("global_load_async_to_lds_b128 %0, %1, %2 offset:4096"
                 :: "v"(l + 4096u), "v"(voff), "s"(gsrc) : "memory");
}

__device__ inline void wait_async_lds() {
    asm volatile("s_wait_asynccnt 0" ::: "memory");
}

// ---------------------------------------------------------------------------
// K=128 WMMA accumulation: acc += A(16x128, LDS) @ L[krow0:krow0+128, :] tile
// A-frag (16x4 f32): lanes0-15 -> K = k+{0,1}; lanes16-31 -> K = k+{2,3}; row = lane&15
// B-frag (4x16 f32): mirrored layout, col = lane&15
// ---------------------------------------------------------------------------
__device__ inline v8f wmma_k128(const float (*A)[CC], const float* __restrict__ L,
                                int krow0, int n0, int lr, int half, v8f acc) {
#pragma unroll 8
    for (int k = 0; k < 128; k += 4) {
        v2f a, b;
        a[0] = A[lr][k + 2 * half + 0];
        a[1] = A[lr][k + 2 * half + 1];
        b[0] = L[(krow0 + k + 2 * half + 0) * CC + n0 + lr];
        b[1] = L[(krow0 + k + 2 * half + 1) * CC + n0 + lr];
        acc = __builtin_amdgcn_wmma_f32_16x16x4_f32(false, a, false, b, (short)0, acc,
                                                    false, false);
    }
    return acc;
}

__global__ void zero_kernel(float* __restrict__ p, int n) {
    int i = blockIdx.x * 256 + threadIdx.x;
    if (i < n) p[i] = 0.0f;
}

// ---------------------------------------------------------------------------
// GCN step: per (batch, gate): G = aggregate(x_t @ W) + bias
// aggregate(h)[d] = dinv[d] * sum_{s<=d} h[s]*dinv[s]   (lower-tri + self loops)
// ---------------------------------------------------------------------------
__global__ __launch_bounds__(256) void gcn_step_kernel(
    const float* __restrict__ x_all, int t,
    const float* __restrict__ Wz, const float* __restrict__ bz,
    const float* __restrict__ Wr, const float* __restrict__ br,
    const float* __restrict__ Wh, const float* __restrict__ bh,
    float* __restrict__ Gz, float* __restrict__ Gr, float* __restrict__ Gh) {
    const int b = blockIdx.x;
    const int g = blockIdx.y;
    const float* W    = (g == 0) ? Wz : (g == 1) ? Wr : Wh;
    const float* bias = (g == 0) ? bz : (g == 1) ? br : bh;
    float*       Gout = (g == 0) ? Gz : (g == 1) ? Gr : Gh;

    __shared__ alignas(16) float sX[NPAD][FIN];   // 12 KB
    __shared__ alignas(16) float sG[NPAD][CC];    // 48 KB

    const int tid = threadIdx.x;
    for (int i = tid; i < NPAD * FIN; i += 256) {
        int n = i >> 5, f = i & 31;
        sX[n][f] = (n < NN) ? x_all[(((size_t)b * TT + t) * NN + n) * FIN + f] : 0.0f;
    }
    __syncthreads();

    const int wave = tid >> 5, lane = tid & 31, half = lane >> 4, lr = lane & 15;
    // 6 M-tiles x 8 N-tiles; wave w owns N-tile w across all M-tiles
    for (int q = wave; q < 48; q += 8) {
        int m0 = (q >> 3) * 16, n0 = (q & 7) * 16;
        v8f acc = {0.f, 0.f, 0.f, 0.f, 0.f, 0.f, 0.f, 0.f};
#pragma unroll
        for (int k = 0; k < FIN; k += 4) {
            v2f a, bb;
            a[0]  = sX[m0 + lr][k + 2 * half + 0];
            a[1]  = sX[m0 + lr][k + 2 * half + 1];
            bb[0] = W[(k + 2 * half + 0) * CC + n0 + lr];
            bb[1] = W[(k + 2 * half + 1) * CC + n0 + lr];
            acc = __builtin_amdgcn_wmma_f32_16x16x4_f32(false, a, false, bb, (short)0,
                                                        acc, false, false);
        }
#pragma unroll
        for (int r = 0; r < 8; ++r)
            sG[m0 + r + 8 * half][n0 + lr] = acc[r];
    }
    __syncthreads();

    if (tid < CC) {
        const int c = tid;
        float acc = 0.0f;
        for (int d = 0; d < NN; ++d) {
            float di = rsqrtf((float)(d + 1));
            acc += sG[d][c] * di;
            Gout[((size_t)b * NPAD + d) * CC + c] = acc * di + bias[c];
        }
        for (int d = NN; d < NPAD; ++d)
            Gout[((size_t)b * NPAD + d) * CC + c] = 0.0f;
    }
}

// ---------------------------------------------------------------------------
// GRU gate step: per (batch, M-tile of 16 rows):
//   Z = sig(Gz@Lz_top + H@Lz_bot + Lzb); R = sig(Gr@Lr_top + H@Lr_bot + Lrb)
//   Ht = tanh(Gh@Lh_top + (H*R)@Lh_bot + Lhb); H = Z*H + (1-Z)*Ht
// Input tiles staged via async global->LDS (ASYNCcnt).
// ---------------------------------------------------------------------------
__global__ __launch_bounds__(256) void gate_step_kernel(
    const float* __restrict__ Gz, const float* __restrict__ Gr,
    const float* __restrict__ Gh,
    const float* __restrict__ LzW, const float* __restrict__ Lzb,
    const float* __restrict__ LrW, const float* __restrict__ Lrb,
    const float* __restrict__ LhW, const float* __restrict__ Lhb,
    float* __restrict__ H) {
    const int b = blockIdx.x, m0 = blockIdx.y * 16;
    __shared__ alignas(16) float sH[16][CC];    // 8 KB each: 48 KB total
    __shared__ alignas(16) float sGz[16][CC];
    __shared__ alignas(16) float sGr[16][CC];
    __shared__ alignas(16) float sGh[16][CC];
    __shared__ alignas(16) float sR[16][CC];
    __shared__ alignas(16) float sHR[16][CC];

    const int tid = threadIdx.x;
    const size_t base = ((size_t)b * NPAD + m0) * CC;

    async_tile_load(H  + base, &sH[0][0],  tid);
    async_tile_load(Gz + base, &sGz[0][0], tid);
    async_tile_load(Gr + base, &sGr[0][0], tid);
    async_tile_load(Gh + base, &sGh[0][0], tid);
    wait_async_lds();
    __syncthreads();

    const int wave = tid >> 5, lane = tid & 31, half = lane >> 4, lr = lane & 15;
    const int n0 = wave * 16;   // 8 waves -> 8 N-tiles of the 128 output channels

    // ---- Z and R tiles ----
    float zb = Lzb[n0 + lr], rb = Lrb[n0 + lr];
    v8f accz = {zb, zb, zb, zb, zb, zb, zb, zb};
    v8f accr = {rb, rb, rb, rb, rb, rb, rb, rb};
    accz = wmma_k128(sGz, LzW, 0,   n0, lr, half, accz);
    accz = wmma_k128(sH,  LzW, 128, n0, lr, half, accz);
    accr = wmma_k128(sGr, LrW, 0,   n0, lr, half, accr);
    accr = wmma_k128(sH,  LrW, 128, n0, lr, half, accr);

    float zv[8];
#pragma unroll
    for (int r = 0; r < 8; ++r) {
        zv[r]    = 1.0f / (1.0f + __expf(-accz[r]));
        float rv = 1.0f / (1.0f + __expf(-accr[r]));
        sR[r + 8 * half][n0 + lr] = rv;
    }
    __syncthreads();

    for (int i = tid; i < 16 * CC; i += 256)
        (&sHR[0][0])[i] = (&sH[0][0])[i] * (&sR[0][0])[i];
    __syncthreads();

    // ---- Ht tile and state update ----
    float hb = Lhb[n0 + lr];
    v8f acch = {hb, hb, hb, hb, hb, hb, hb, hb};
    acch = wmma_k128(sGh, LhW, 0,   n0, lr, half, acch);
    acch = wmma_k128(sHR, LhW, 128, n0, lr, half, acch);

#pragma unroll
    for (int r = 0; r < 8; ++r) {
        int m = r + 8 * half;
        float ht   = tanhf(acch[r]);
        float hold = sH[m][n0 + lr];
        float hn   = zv[r] * hold + (1.0f - zv[r]) * ht;
        if (m0 + m < NN) H[base + (size_t)m * CC + n0 + lr] = hn;
    }
}

// ---------------------------------------------------------------------------
// Head: pooled = mean_n(relu(H)) @ Wout + bout, then concat extras
// ---------------------------------------------------------------------------
__global__ __launch_bounds__(128) void head_kernel(
    const float* __restrict__ H, const float* __restrict__ Wout,
    const float* __restrict__ bout, const float* __restrict__ hide,
    const float* __restrict__ tstep, const float* __restrict__ lastk,
    float* __restrict__ out) {
    const int b = blockIdx.x, tid = threadIdx.x;
    __shared__ float cs[CC];
    float s = 0.0f;
    for (int n = 0; n < NN; ++n)
        s += fmaxf(H[((size_t)b * NPAD + n) * CC + tid], 0.0f);
    cs[tid] = s;
    __syncthreads();
    const int OUTW = 16 + 2 + 1 + 24;  // 43
    if (tid < 16) {
        float a = 0.0f;
        for (int c = 0; c < CC; ++c) a += cs[c] * Wout[c * 16 + tid];
        out[b * OUTW + tid] = a * (1.0f / (float)NN) + bout[tid];
    }
    if (tid < 2)  out[b * OUTW + 16 + tid] = hide[b * 2 + tid];
    if (tid == 0) out[b * OUTW + 18]       = tstep[b];
    if (tid < 24) out[b * OUTW + 19 + tid] = lastk[b * 24 + tid];
}

extern "C" void kernel_launch(void* const* d_in, const int* in_sizes, int n_in,
                              void* d_out, int out_size, void* d_ws, size_t ws_size,
                              hipStream_t stream) {
    const float* agent = (const float*)d_in[0];
    const float* hide  = (const float*)d_in[1];
    const float* tstep = (const float*)d_in[2];
    const float* lastk = (const float*)d_in[3];
    const float* Wz  = (const float*)d_in[4];
    const float* bz  = (const float*)d_in[5];
    const float* Wr  = (const float*)d_in[6];
    const float* br  = (const float*)d_in[7];
    const float* Wh  = (const float*)d_in[8];
    const float* bh  = (const float*)d_in[9];
    const float* LzW = (const float*)d_in[10];
    const float* Lzb = (const float*)d_in[11];
    const float* LrW = (const float*)d_in[12];
    const float* Lrb = (const float*)d_in[13];
    const float* LhW = (const float*)d_in[14];
    const float* Lhb = (const float*)d_in[15];
    const float* Wout = (const float*)d_in[16];
    const float* bout = (const float*)d_in[17];
    float* out = (float*)d_out;

    float* H  = (float*)d_ws;         // [B, NPAD, C]
    float* Gz = H + HSZ;
    float* Gr = Gz + HSZ;
    float* Gh = Gr + HSZ;

    zero_kernel<<<(HSZ + 255) / 256, 256, 0, stream>>>(H, HSZ);

    for (int t = 0; t < TT; ++t) {
        gcn_step_kernel<<<dim3(BB, 3), 256, 0, stream>>>(
            agent, t, Wz, bz, Wr, br, Wh, bh, Gz, Gr, Gh);
        gate_step_kernel<<<dim3(BB, 6), 256, 0, stream>>>(
            Gz, Gr, Gh, LzW, Lzb, LrW, Lrb, LhW, Lhb, H);
    }

    head_kernel<<<BB, 128, 0, stream>>>(H, Wout, bout, hide, tstep, lastk, out);
}